// GraphEncoder_90374701842554
// MI455X (gfx1250) — compile-verified
//
#include <hip/hip_runtime.h>

typedef float v2f __attribute__((ext_vector_type(2)));
typedef float v8f __attribute__((ext_vector_type(8)));

// ---------------------------------------------------------------------------
// Native fp32 global atomic add (global_atomic_add_f32). unsafeAtomicAdd is
// HIP's header-provided path that attaches the no-fine-grained-memory
// metadata so the backend selects the hardware fadd instead of a CAS loop.
// ---------------------------------------------------------------------------
__device__ __forceinline__ void atomic_add_f32(float* p, float v) {
  unsafeAtomicAdd(p, v);
}

// ---------------------------------------------------------------------------
// Degree / normalization
// ---------------------------------------------------------------------------
__global__ void deg_init_kernel(float* deg, int n) {
  int i = blockIdx.x * blockDim.x + threadIdx.x;
  if (i < n) deg[i] = 1.0f;              // self-loop contributes 1
}

__global__ void deg_edge_kernel(float* deg, const int* __restrict__ dst, int e) {
  int i = blockIdx.x * blockDim.x + threadIdx.x;
  if (i < e) atomic_add_f32(&deg[dst[i]], 1.0f);
}

__global__ void inv_sqrt_kernel(float* deg, int n) {
  int i = blockIdx.x * blockDim.x + threadIdx.x;
  if (i < n) {
    float d = deg[i];                    // always >= 1
    deg[i] = __frsqrt_rn(d);
  }
}

// ---------------------------------------------------------------------------
// FP32 WMMA GEMM: H[nrows x F] = X[nrows x K] @ W[K x F], K and F compile-time
// so every global load in the unrolled loop is base + immediate offset.
// One wave (32 lanes) computes one 16x16 output tile via K/4 x 16x16x4 WMMA.
//
// A (16x4 f32): lanes 0-15 hold M=lane, VGPR0=K0, VGPR1=K1;
//               lanes 16-31 hold M=lane-16, VGPR0=K2, VGPR1=K3.
// B (4x16 f32): lanes 0-15 hold N=lane, VGPR0=K0, VGPR1=K1;
//               lanes 16-31 hold N=lane-16, VGPR0=K2, VGPR1=K3.
// C/D (16x16 f32): VGPR r: lanes 0-15 -> M=r, lanes 16-31 -> M=r+8.
// ---------------------------------------------------------------------------
template <int K, int F>
__global__ void gcn_gemm_wmma(const float* __restrict__ X,
                              const float* __restrict__ W,
                              float* __restrict__ H,
                              int nrows) {
  const int lane = threadIdx.x & 31;
  const int wave = threadIdx.x >> 5;
  const int waves_per_block = blockDim.x >> 5;
  const int tiles_m = nrows >> 4;

  const int tm = blockIdx.x * waves_per_block + wave;   // tile row (wave-uniform)
  const int tn = blockIdx.y;                            // tile col
  if (tm >= tiles_m) return;                            // whole wave exits together

  const int half = lane >> 4;            // 0: lanes 0-15, 1: lanes 16-31
  const int l15  = lane & 15;
  const int row  = tm * 16 + l15;        // A row for this lane
  const int col  = tn * 16 + l15;        // B/D column for this lane
  const int koff = half * 2;             // K sub-offset within the 4-chunk

  const float* __restrict__ xrow  = X + (long long)row * K + koff;  // + kk at use
  const float* __restrict__ wbase = W + (long long)koff * F + col;  // + kk*F at use

  v8f c = {};
#pragma unroll
  for (int kk = 0; kk < K; kk += 4) {
    v2f a = *(const v2f*)(xrow + kk);    // K = kk+koff, kk+koff+1
    v2f b;
    b[0] = wbase[kk * F];                // row kk+koff   -> immediate offset
    b[1] = wbase[kk * F + F];            // row kk+koff+1 -> immediate offset
    c = __builtin_amdgcn_wmma_f32_16x16x4_f32(
        /*neg_a=*/false, a, /*neg_b=*/false, b,
        /*c_mod=*/(short)0, c, /*reuse_a=*/false, /*reuse_b=*/false);
  }

  float* __restrict__ hbase = H + (long long)(tm * 16 + half * 8) * F + col;
#pragma unroll
  for (int r = 0; r < 8; ++r)
    hbase[r * F] = c[r];
}

// ---------------------------------------------------------------------------
// Zero an fp32 buffer (float4 vectorized; count is a multiple of 4)
// ---------------------------------------------------------------------------
__global__ void zero_kernel(float4* p, long long count4) {
  long long i = (long long)blockIdx.x * blockDim.x + threadIdx.x;
  if (i < count4) p[i] = make_float4(0.f, 0.f, 0.f, 0.f);
}

// ---------------------------------------------------------------------------
// Edge scatter: ACC[dst] += inv[src]*inv[dst] * H[src], one wave per edge.
// ---------------------------------------------------------------------------
__global__ void gcn_scatter(const float* __restrict__ H,
                            const float* __restrict__ inv,
                            const int* __restrict__ src,
                            const int* __restrict__ dst,
                            float* __restrict__ ACC,
                            int e, int F) {
  const int warp = (int)(((long long)blockIdx.x * blockDim.x + threadIdx.x) >> 5);
  const int lane = threadIdx.x & 31;
  if (warp >= e) return;
  const int s = src[warp];
  const int d = dst[warp];
  const float nm = inv[s] * inv[d];
  const float* __restrict__ hs = H + (long long)s * F;
  float* __restrict__ ad = ACC + (long long)d * F;
  if (F == 64) {                         // fast path: one float2 per lane
    v2f h2 = *(const v2f*)(hs + lane * 2);
    atomic_add_f32(ad + lane * 2 + 0, h2[0] * nm);
    atomic_add_f32(ad + lane * 2 + 1, h2[1] * nm);
  } else {
    for (int f = lane; f < F; f += 32)
      atomic_add_f32(ad + f, hs[f] * nm);
  }
}

// ---------------------------------------------------------------------------
// Finalize: OUT = [relu]( ACC + inv[n]^2 * H + b ).  OUT may alias ACC.
// float4 vectorized: F is a multiple of 4, rows are 16B-aligned.
// ---------------------------------------------------------------------------
__global__ void gcn_finalize(const float4* __restrict__ ACC,
                             const float4* __restrict__ H,
                             const float* __restrict__ inv,
                             const float4* __restrict__ b,
                             float4* __restrict__ OUT,
                             int nodes, int F, int do_relu) {
  long long idx = (long long)blockIdx.x * blockDim.x + threadIdx.x;
  const int F4 = F >> 2;
  long long total4 = (long long)nodes * F4;
  if (idx >= total4) return;
  int n  = (int)(idx / F4);
  int f4 = (int)(idx % F4);
  float iv = inv[n];
  float iv2 = iv * iv;
  float4 acc = ACC[idx];
  float4 h   = H[idx];
  float4 bb  = b[f4];
  float4 v;
  v.x = acc.x + iv2 * h.x + bb.x;
  v.y = acc.y + iv2 * h.y + bb.y;
  v.z = acc.z + iv2 * h.z + bb.z;
  v.w = acc.w + iv2 * h.w + bb.w;
  if (do_relu) {
    v.x = fmaxf(v.x, 0.f); v.y = fmaxf(v.y, 0.f);
    v.z = fmaxf(v.z, 0.f); v.w = fmaxf(v.w, 0.f);
  }
  OUT[idx] = v;
}

// ---------------------------------------------------------------------------
// Host side
// ---------------------------------------------------------------------------
static inline int ceil_div(long long a, int b) { return (int)((a + b - 1) / b); }

template <int F>
static void run_gemm64(const float* X, const float* W, float* H,
                       int nrows, hipStream_t stream) {
  const int waves_per_block = 4;
  const int tiles_m = nrows / 16;
  dim3 grid(ceil_div(tiles_m, waves_per_block), F / 16);
  dim3 block(32 * waves_per_block);
  gcn_gemm_wmma<64, F><<<grid, block, 0, stream>>>(X, W, H, nrows);
}

static void run_propagate(const float* Hsrc, const float* inv,
                          const int* src, const int* dst,
                          const float* bias, float* ACC_and_OUT,
                          int nodes, int e, int F, int do_relu,
                          hipStream_t stream) {
  long long total4 = (long long)nodes * (F >> 2);
  zero_kernel<<<ceil_div(total4, 256), 256, 0, stream>>>((float4*)ACC_and_OUT, total4);
  // one wave per edge -> e*32 threads
  long long sthreads = (long long)e * 32;
  gcn_scatter<<<ceil_div(sthreads, 256), 256, 0, stream>>>(Hsrc, inv, src, dst,
                                                           ACC_and_OUT, e, F);
  gcn_finalize<<<ceil_div(total4, 256), 256, 0, stream>>>(
      (const float4*)ACC_and_OUT, (const float4*)Hsrc, inv, (const float4*)bias,
      (float4*)ACC_and_OUT, nodes, F, do_relu);
}

extern "C" void kernel_launch(void* const* d_in, const int* in_sizes, int n_in,
                              void* d_out, int out_size, void* d_ws, size_t ws_size,
                              hipStream_t stream) {
  const float* x   = (const float*)d_in[0];
  const int*   ei  = (const int*)  d_in[1];
  const float* W1  = (const float*)d_in[2];
  const float* b1  = (const float*)d_in[3];
  const float* W2  = (const float*)d_in[4];
  const float* b2  = (const float*)d_in[5];
  const float* Wmu = (const float*)d_in[6];
  const float* bmu = (const float*)d_in[7];
  const float* Wlv = (const float*)d_in[8];
  const float* blv = (const float*)d_in[9];

  const int D_H = 64, D_L = 32;
  const int n = in_sizes[0] / 64;         // 100000
  const int e = in_sizes[1] / 2;          // 1000000
  const int* src = ei;                    // edge_index[0, :]
  const int* dst = ei + e;                // edge_index[1, :]

  // Workspace layout (floats): inv[n] | Htmp[n*64] | Hbuf[n*64]
  float* ws   = (float*)d_ws;
  float* inv  = ws;
  float* Htmp = ws + n;
  float* Hbuf = Htmp + (long long)n * D_H;

  float* mu = (float*)d_out;
  float* lv = mu + (long long)n * D_L;

  // --- degree + symmetric normalization ---
  deg_init_kernel<<<ceil_div(n, 256), 256, 0, stream>>>(inv, n);
  deg_edge_kernel<<<ceil_div(e, 256), 256, 0, stream>>>(inv, dst, e);
  inv_sqrt_kernel<<<ceil_div(n, 256), 256, 0, stream>>>(inv, n);

  // --- layer 1: Hbuf = relu( prop(x @ W1) + b1 ) ---
  run_gemm64<64>(x, W1, Htmp, n, stream);
  run_propagate(Htmp, inv, src, dst, b1, Hbuf, n, e, D_H, /*relu=*/1, stream);

  // --- layer 2: Hbuf = relu( prop(Hbuf @ W2) + b2 ) ---
  run_gemm64<64>(Hbuf, W2, Htmp, n, stream);
  run_propagate(Htmp, inv, src, dst, b2, Hbuf, n, e, D_H, /*relu=*/1, stream);

  // --- mu head: d_out[0 : n*32] ---
  run_gemm64<32>(Hbuf, Wmu, Htmp, n, stream);   // Htmp[:n*32] used
  run_propagate(Htmp, inv, src, dst, bmu, mu, n, e, D_L, /*relu=*/0, stream);

  // --- logvar head: d_out[n*32 : 2*n*32] ---
  run_gemm64<32>(Hbuf, Wlv, Htmp, n, stream);
  run_propagate(Htmp, inv, src, dst, blv, lv, n, e, D_L, /*relu=*/0, stream);
}